// PixelTransformer_39805756899386
// MI455X (gfx1250) — compile-verified
//
#include <hip/hip_runtime.h>
#include <math.h>

// ---------------------------------------------------------------------------
// PixelTransformer normalizing-flow NLL for MI455X (gfx1250), fp32 WMMA path.
//
// Key exact simplification: h never depends on x (tokens are positional
// only), so attention over the batch axis collapses: softmax of a constant
// row is uniform and o == v. The transformer is a per-token MLP stack.
// ---------------------------------------------------------------------------

typedef float v2f __attribute__((ext_vector_type(2)));
typedef float v8f __attribute__((ext_vector_type(8)));

#define LAYERS 8
#define DMODEL 5
#define FFDIM  2048
#define NTOK   1024
#define HPAD   8          // D padded to 8 (two K=4 WMMA steps)
#define BATCH  32
#define EPSV   1e-5f
#define HALF_LOG_2PI 0.91893853320467274178f

__device__ __forceinline__ v8f wmma_f32_16x16x4(v2f a, v2f b, v8f c) {
  // 8-arg pattern: (neg_a, A, neg_b, B, c_mod, C, reuse_a, reuse_b)
  return __builtin_amdgcn_wmma_f32_16x16x4_f32(false, a, false, b,
                                               (short)0, c, false, false);
}

// --- 1. positional tokens -> hbuf[1024][8] (cols 5..7 zero) ---------------
__global__ void init_tok(float* __restrict__ hbuf) {
  int n = blockIdx.x * blockDim.x + threadIdx.x;
  if (n >= NTOK) return;
  int xc = n & 31;          // column (fast axis of (H,W))
  int yc = n >> 5;          // row
  float xs = (float)xc * 1e-4f;
  float ys = (float)yc * 1e-4f;
  float* h = hbuf + n * HPAD;
  h[0] = -1.0f;
  h[1] = sinf(xs);
  h[2] = cosf(xs);
  h[3] = sinf(ys);
  h[4] = cosf(ys);
  h[5] = 0.0f; h[6] = 0.0f; h[7] = 0.0f;
}

// --- 2. one transformer layer: one wave (32 threads) per 16-token tile ----
__global__ void __launch_bounds__(32)
tf_layer(float* __restrict__ hbuf,
         const float* __restrict__ wqkv, const float* __restrict__ bqkv,
         const float* __restrict__ wo,   const float* __restrict__ bo,
         const float* __restrict__ w1,   const float* __restrict__ b1,
         const float* __restrict__ w2,   const float* __restrict__ b2,
         const float* __restrict__ ln1w, const float* __restrict__ ln1b,
         const float* __restrict__ ln2w, const float* __restrict__ ln2b,
         int l) {
  __shared__ float T0[16][HPAD];   // LN1 output, A-staging for GEMM1
  __shared__ float T1[16][17];     // relu tile / final FFN out (padded stride)

  const int lane = threadIdx.x;    // 32-thread block == 1 wave32
  const int row  = lane & 15;      // M (A/C) or N (B) index
  const int hi   = lane >> 4;      // lane half selects K pair
  const int m0   = blockIdx.x * 16;

  const float* wqkv_l = wqkv + l * 3 * DMODEL * DMODEL;
  const float* bqkv_l = bqkv + l * 3 * DMODEL;
  const float* wo_l   = wo   + l * DMODEL * DMODEL;
  const float* bo_l   = bo   + l * DMODEL;
  const float* w1_l   = w1   + l * FFDIM * DMODEL;
  const float* b1_l   = b1   + l * FFDIM;
  const float* w2_l   = w2   + l * DMODEL * FFDIM;
  const float* b2_l   = b2   + l * DMODEL;
  const float* ln1w_l = ln1w + l * DMODEL;
  const float* ln1b_l = ln1b + l * DMODEL;
  const float* ln2w_l = ln2w + l * DMODEL;
  const float* ln2b_l = ln2b + l * DMODEL;

  float hm[DMODEL];                // LN1 output, owned by lanes 0..15

  // ---- per-token scalar stage: qkv -> (attention == v) -> Wo -> LN1 ----
  if (lane < 16) {
    const int m = m0 + lane;
    float hv[DMODEL];
#pragma unroll
    for (int d = 0; d < DMODEL; ++d) hv[d] = hbuf[m * HPAD + d];

    // Only v (rows 2D..3D-1 of wqkv) survives the collapsed attention.
    float vv[DMODEL];
#pragma unroll
    for (int j = 0; j < DMODEL; ++j) {
      float acc = bqkv_l[2 * DMODEL + j];
#pragma unroll
      for (int d = 0; d < DMODEL; ++d)
        acc += wqkv_l[(2 * DMODEL + j) * DMODEL + d] * hv[d];
      vv[j] = acc;
    }
    float r[DMODEL];
    float mean = 0.0f;
#pragma unroll
    for (int i = 0; i < DMODEL; ++i) {
      float acc = bo_l[i];
#pragma unroll
      for (int j = 0; j < DMODEL; ++j) acc += wo_l[i * DMODEL + j] * vv[j];
      r[i] = hv[i] + acc;
      mean += r[i];
    }
    mean *= 0.2f;
    float var = 0.0f;
#pragma unroll
    for (int i = 0; i < DMODEL; ++i) {
      float d0 = r[i] - mean;
      var += d0 * d0;
    }
    var *= 0.2f;
    float inv = rsqrtf(var + EPSV);
#pragma unroll
    for (int i = 0; i < DMODEL; ++i) {
      hm[i] = (r[i] - mean) * inv * ln1w_l[i] + ln1b_l[i];
      T0[lane][i] = hm[i];
    }
    T0[lane][5] = 0.0f; T0[lane][6] = 0.0f; T0[lane][7] = 0.0f;
  }
  __syncthreads();

  // ---- A operands for GEMM1: 16x8 in two 16x4 tiles (ISA A striping:
  //      lanes hold M=row; VGPR/lane-half hold K = hi*2 + {0,1}) ----
  v2f a0, a1;
  a0.x = T0[row][hi * 2 + 0];
  a0.y = T0[row][hi * 2 + 1];
  a1.x = T0[row][4 + hi * 2 + 0];
  a1.y = T0[row][4 + hi * 2 + 1];

  // Masks instead of conditional loads: keeps the FF loop free of
  // EXEC-divergent branches (no s_and_saveexec in the hot loop).
  const float mhi  = hi ? 0.0f : 1.0f;            // K=4 real, K=6 pad
  const float mrow = (row < DMODEL) ? 1.0f : 0.0f; // W2 rows 5..15 are pad
  const int   rcl  = (row < DMODEL) ? row : 0;     // clamped (load always legal)
  const float* w2_r = w2_l + rcl * FFDIM;          // 8KB-aligned row base

  v8f acc2 = {};                    // GEMM2 accumulator (16 tokens x 16 cols)

  for (int ff0 = 0; ff0 < FFDIM; ff0 += 16) {
    const int f = ff0 + row;        // B-matrix N index for GEMM1

    if (ff0 + 16 < FFDIM)
      __builtin_prefetch(&w1_l[(ff0 + 16) * DMODEL], 0, 3);

    // GEMM1: C1[m,f] = sum_d H[m,d] * W1[f,d]   (B striping mirrors A:
    // lanes hold N=f; VGPR/half hold K = hi*2 + {0,1})
    v2f b0, bb1;
    b0.x  = w1_l[f * DMODEL + (hi * 2 + 0)];
    b0.y  = w1_l[f * DMODEL + (hi * 2 + 1)];
    bb1.x = w1_l[f * DMODEL + 4] * mhi;         // unconditional load, masked
    bb1.y = 0.0f;                               // K=5,7 pad

    v8f c1 = {};
    c1 = wmma_f32_16x16x4(a0, b0, c1);
    c1 = wmma_f32_16x16x4(a1, bb1, c1);

    // bias + relu, redistribute D-layout -> LDS tile T1[m][f_local]
    const float bias = b1_l[f];
#pragma unroll
    for (int rr = 0; rr < 8; ++rr)
      T1[rr + hi * 8][row] = fmaxf(c1[rr] + bias, 0.0f);
    __syncthreads();

    // GEMM2: C2[m,i] += sum_k T1[m,k] * W2[i, ff0+k], 4 k-steps of 4
#pragma unroll
    for (int j = 0; j < 4; ++j) {
      v2f a;
      a.x = T1[row][4 * j + hi * 2 + 0];
      a.y = T1[row][4 * j + hi * 2 + 1];
      const int kk = ff0 + 4 * j + hi * 2;      // even -> 8B aligned pair
      v2f b = *(const v2f*)(w2_r + kk);
      b.x *= mrow;
      b.y *= mrow;
      acc2 = wmma_f32_16x16x4(a, b, acc2);
    }
    __syncthreads();                // before next iteration rewrites T1
  }

  // ---- spill FFN output (cols 0..4) and finish: bias, residual, LN2 ----
  if (row < DMODEL) {
#pragma unroll
    for (int rr = 0; rr < 8; ++rr)
      T1[rr + hi * 8][row] = acc2[rr];
  }
  __syncthreads();

  if (lane < 16) {
    const int m = m0 + lane;
    float r2[DMODEL];
    float mean = 0.0f;
#pragma unroll
    for (int d = 0; d < DMODEL; ++d) {
      r2[d] = hm[d] + (T1[lane][d] + b2_l[d]);
      mean += r2[d];
    }
    mean *= 0.2f;
    float var = 0.0f;
#pragma unroll
    for (int d = 0; d < DMODEL; ++d) {
      float d0 = r2[d] - mean;
      var += d0 * d0;
    }
    var *= 0.2f;
    float inv = rsqrtf(var + EPSV);
#pragma unroll
    for (int d = 0; d < DMODEL; ++d)
      hbuf[m * HPAD + d] = (r2[d] - mean) * inv * ln2w_l[d] + ln2b_l[d];
  }
}

// --- 3. flow head: per-token scales/shifts + partial sum of raw s_ --------
__global__ void __launch_bounds__(256)
head_kernel(const float* __restrict__ hbuf,
            const float* __restrict__ f0w1, const float* __restrict__ f0b1,
            const float* __restrict__ f0w2, const float* __restrict__ f0b2,
            const float* __restrict__ sfac,
            float* __restrict__ scales, float* __restrict__ tsh,
            float* __restrict__ pS) {
  __shared__ float red[256];
  const int tid = threadIdx.x;
  const int n = blockIdx.x * 256 + tid;           // n < 1024 by grid sizing

  float hv[DMODEL];
#pragma unroll
  for (int d = 0; d < DMODEL; ++d) hv[d] = hbuf[n * HPAD + d];

  float u[16];
#pragma unroll
  for (int i = 0; i < 16; ++i) {
    float acc = f0b1[i];
#pragma unroll
    for (int d = 0; d < DMODEL; ++d) acc += f0w1[i * DMODEL + d] * hv[d];
    u[i] = fmaxf(acc, 0.0f);
  }

  const float sf = expf(sfac[0]);
  float ssum = 0.0f;
  for (int i = 0; i < 16; ++i) {
    float s = f0b2[i];
    float t = f0b2[16 + i];
#pragma unroll
    for (int j = 0; j < 16; ++j) {
      s += f0w2[i * 16 + j] * u[j];
      t += f0w2[(16 + i) * 16 + j] * u[j];
    }
    scales[n * 16 + i] = tanhf(s / sf) * sf;
    tsh[n * 16 + i] = t;
    ssum += s;                                     // sldj uses raw s_
  }

  red[tid] = ssum;
  __syncthreads();
  for (int stride = 128; stride > 0; stride >>= 1) {
    if (tid < stride) red[tid] += red[tid + stride];
    __syncthreads();
  }
  if (tid == 0) pS[blockIdx.x] = red[0];
}

// --- 4. affine flow over (B,N) + partial Gaussian log-density sums --------
__global__ void __launch_bounds__(256)
flow_kernel(const float* __restrict__ x,
            const float* __restrict__ scales, const float* __restrict__ tsh,
            float* __restrict__ pZ) {
  __shared__ float red[256];
  const int tid = threadIdx.x;
  const int idx = blockIdx.x * 256 + tid;          // < 32768
  const int n = idx & (NTOK - 1);

  float z = x[idx];                                // x[b*1024 + n]
  for (int i = 0; i < 16; ++i)
    z = expf(scales[n * 16 + i]) * z + tsh[n * 16 + i];

  red[tid] = -0.5f * z * z - HALF_LOG_2PI;
  __syncthreads();
  for (int stride = 128; stride > 0; stride >>= 1) {
    if (tid < stride) red[tid] += red[tid + stride];
    __syncthreads();
  }
  if (tid == 0) pZ[blockIdx.x] = red[0];
}

// --- 5. deterministic finalize: out = -(B * sum_s + sum_z) ----------------
__global__ void finalize_kernel(const float* __restrict__ pS,
                                const float* __restrict__ pZ,
                                float* __restrict__ out) {
  if (threadIdx.x == 0 && blockIdx.x == 0) {
    float ssum = 0.0f;
    for (int i = 0; i < 4; ++i) ssum += pS[i];
    float zsum = 0.0f;
    for (int i = 0; i < 128; ++i) zsum += pZ[i];
    out[0] = -((float)BATCH * ssum + zsum);
  }
}

extern "C" void kernel_launch(void* const* d_in, const int* in_sizes, int n_in,
                              void* d_out, int out_size, void* d_ws, size_t ws_size,
                              hipStream_t stream) {
  const float* x    = (const float*)d_in[0];
  const float* wqkv = (const float*)d_in[1];
  const float* bqkv = (const float*)d_in[2];
  const float* wo   = (const float*)d_in[3];
  const float* bo   = (const float*)d_in[4];
  const float* w1   = (const float*)d_in[5];
  const float* b1   = (const float*)d_in[6];
  const float* w2   = (const float*)d_in[7];
  const float* b2   = (const float*)d_in[8];
  const float* ln1w = (const float*)d_in[9];
  const float* ln1b = (const float*)d_in[10];
  const float* ln2w = (const float*)d_in[11];
  const float* ln2b = (const float*)d_in[12];
  const float* f0w1 = (const float*)d_in[13];
  const float* f0b1 = (const float*)d_in[14];
  const float* f0w2 = (const float*)d_in[15];
  const float* f0b2 = (const float*)d_in[16];
  const float* sfac = (const float*)d_in[17];

  float* ws     = (float*)d_ws;
  float* hbuf   = ws;                    // 1024*8
  float* scales = ws + NTOK * HPAD;      // 1024*16
  float* tsh    = scales + NTOK * 16;    // 1024*16
  float* pS     = tsh + NTOK * 16;       // 4
  float* pZ     = pS + 4;                // 128

  init_tok<<<4, 256, 0, stream>>>(hbuf);
  for (int l = 0; l < LAYERS; ++l)
    tf_layer<<<NTOK / 16, 32, 0, stream>>>(hbuf, wqkv, bqkv, wo, bo,
                                           w1, b1, w2, b2,
                                           ln1w, ln1b, ln2w, ln2b, l);
  head_kernel<<<4, 256, 0, stream>>>(hbuf, f0w1, f0b1, f0w2, f0b2, sfac,
                                     scales, tsh, pS);
  flow_kernel<<<(BATCH * NTOK) / 256, 256, 0, stream>>>(x, scales, tsh, pZ);
  finalize_kernel<<<1, 32, 0, stream>>>(pS, pZ, (float*)d_out);
}